// FCOSLoss_64226940944786
// MI455X (gfx1250) — compile-verified
//
#include <hip/hip_runtime.h>
#include <math.h>

#define INF_F  1.0e9f
#define ALPHA_F 0.25f

// CDNA5 async global->LDS staging (ASYNCcnt-tracked), if the toolchain has it.
#if defined(__has_builtin)
#if __has_builtin(__builtin_amdgcn_global_load_async_to_lds_b32) && \
    __has_builtin(__builtin_amdgcn_s_wait_asynccnt)
#define USE_ASYNC_LDS 1
#endif
#endif

#define AS_GLOBAL __attribute__((address_space(1)))
#define AS_SHARED __attribute__((address_space(3)))

typedef float v2f __attribute__((ext_vector_type(2)));
typedef float v8f __attribute__((ext_vector_type(8)));

// Exact fp32 wave32 sum using V_WMMA_F32_16X16X4_F32.
// A (16x4): VGPR0 = v (each lane's value occupies exactly one A slot), VGPR1 = 0.
// B (4x16): all ones  =>  D[m,n] = sum_k A[m,k] = v[m] + v[m+16].
// Lane l (l<16) holds D[0..7, l] in its 8 D VGPRs; lane l+16 holds D[8..15, l].
// Sum own 8 VGPRs, then add xor-16 partner => every lane gets the full 32-lane sum.
__device__ __forceinline__ float wave_sum32(float v) {
    v2f a; a[0] = v;    a[1] = 0.0f;
    v2f b; b[0] = 1.0f; b[1] = 1.0f;
    v8f c = {};
    c = __builtin_amdgcn_wmma_f32_16x16x4_f32(
            /*neg_a=*/false, a, /*neg_b=*/false, b,
            /*c_mod=*/(short)0, c, /*reuse_a=*/false, /*reuse_b=*/false);
    float s = c[0] + c[1] + c[2] + c[3] + c[4] + c[5] + c[6] + c[7];
    s += __shfl_xor(s, 16, 32);
    return s;
}

// log_sigmoid(x) and log_sigmoid(-x), numerically stable
__device__ __forceinline__ void log_sig_pair(float x, float& lp, float& lnp) {
    float ax  = fabsf(x);
    float l1p = log1pf(expf(-ax));
    lp  = (x >= 0.0f) ? -l1p : (x - l1p);     // log_sigmoid(x)
    lnp = (x >= 0.0f) ? (-x - l1p) : -l1p;    // log_sigmoid(-x)
}

__global__ void __launch_bounds__(256)
fcos_fused_kernel(const float* __restrict__ loc,       // N*3
                  const float* __restrict__ cls_pred,  // B*N*C
                  const float* __restrict__ box_pred,  // B*N*6
                  const float* __restrict__ ctr_pred,  // B*N
                  const float* __restrict__ bboxes,    // B*G*6
                  const int*   __restrict__ gt_labels, // B*G
                  const float* __restrict__ spt,       // N
                  const float* __restrict__ soi,       // N*2
                  float* __restrict__ partials,        // nblk*5
                  int N, int B, int G, int C)
{
    __shared__ float sBB[64 * 6];
    __shared__ int   sLab[64];
    __shared__ float sRed[8][5];

    const int tid = threadIdx.x;
    const int b   = blockIdx.y;
    const int n   = blockIdx.x * blockDim.x + tid;

    // ---- stage this image's boxes + labels into LDS ----
#ifdef USE_ASYNC_LDS
    {
        // Direct global->LDS async copy: no VGPR round-trip, tracked by ASYNCcnt.
        // Builtin prototype (from compiler diagnostics): (int AS1* src, int AS3* dst, imm, imm).
        AS_GLOBAL int* gbb = (AS_GLOBAL int*)(bboxes + (size_t)b * G * 6);
        AS_SHARED int* lbb = (AS_SHARED int*)sBB;
        for (int i = tid; i < G * 6; i += blockDim.x)
            __builtin_amdgcn_global_load_async_to_lds_b32(gbb + i, lbb + i, 0, 0);
        AS_GLOBAL int* glb = (AS_GLOBAL int*)(gt_labels + (size_t)b * G);
        AS_SHARED int* llb = (AS_SHARED int*)sLab;
        for (int i = tid; i < G; i += blockDim.x)
            __builtin_amdgcn_global_load_async_to_lds_b32(glb + i, llb + i, 0, 0);
        __builtin_amdgcn_s_wait_asynccnt(0);
    }
#else
    for (int i = tid; i < G * 6; i += blockDim.x) sBB[i]  = bboxes[b * G * 6 + i];
    for (int i = tid; i < G;     i += blockDim.x) sLab[i] = gt_labels[b * G + i];
#endif
    __syncthreads();

    float s_cls = 0.0f, s_npos = 0.0f, s_box = 0.0f, s_w = 0.0f, s_ctr = 0.0f;

    if (n < N) {
        const float* cl = cls_pred + ((size_t)b * N + n) * C;
        const float* bp = box_pred + ((size_t)b * N + n) * 6;
        __builtin_prefetch(cl, 0, 0);     // global_prefetch_b8
        __builtin_prefetch(bp, 0, 0);

        const float x = loc[n * 3 + 0];
        const float y = loc[n * 3 + 1];
        const float z = loc[n * 3 + 2];
        const float s  = spt[n];
        const float lo = soi[n * 2 + 0];
        const float hi = soi[n * 2 + 1];

        // ---- target assignment: argmin over boxes of masked area ----
        float best = INF_F;
        int   bg   = 0;
        for (int g = 0; g < G; ++g) {
            const float b0 = sBB[g * 6 + 0], b1 = sBB[g * 6 + 1], b2 = sBB[g * 6 + 2];
            const float b3 = sBB[g * 6 + 3], b4 = sBB[g * 6 + 4], b5 = sBB[g * 6 + 5];
            const float l  = x - b0, t  = y - b1, f  = z - b2;
            const float r  = b3 - x, bo = b4 - y, a  = b5 - z;
            const float mt = fmaxf(fmaxf(fmaxf(l, t), fmaxf(f, r)), fmaxf(bo, a));
            const bool cared = (mt >= lo) && (mt <= hi);
            const float cx = (b0 + b3) * 0.5f, cy = (b1 + b4) * 0.5f, cz = (b2 + b5) * 0.5f;
            const float cminx = fmaxf(cx - s, b0), cminy = fmaxf(cy - s, b1), cminz = fmaxf(cz - s, b2);
            const float cmaxx = fminf(cx + s, b3), cmaxy = fminf(cy + s, b4), cmaxz = fminf(cz + s, b5);
            const float dmin = fminf(fminf(fminf(x - cminx, y - cminy), fminf(z - cminz, cmaxx - x)),
                                     fminf(cmaxy - y, cmaxz - z));
            const bool is_in = dmin > 0.0f;
            const float area = (b3 - b0) * (b4 - b1) * (b5 - b2);
            const float l2a  = (is_in && cared) ? area : INF_F;
            if (l2a < best) { best = l2a; bg = g; }   // strict '<' == argmin first-occurrence
        }
        const int  label = (best == INF_F) ? 0 : sLab[bg];
        const bool pos   = label > 0;

        // ---- focal loss over C classes (all locations) ----
        for (int j = 0; j < C; ++j) {
            const float xv = cl[j];
            const float p  = 1.0f / (1.0f + expf(-xv));
            float lp, lnp;
            log_sig_pair(xv, lp, lnp);
            const float omp = 1.0f - p;
            s_cls += (label == j + 1) ? (-ALPHA_F * omp * omp * lp)
                                      : (-(1.0f - ALPHA_F) * p * p * lnp);
        }

        // ---- box (IoU*centerness) + centerness BCE, positives only ----
        if (pos) {
            s_npos = 1.0f;
            const float b0 = sBB[bg * 6 + 0], b1 = sBB[bg * 6 + 1], b2 = sBB[bg * 6 + 2];
            const float b3 = sBB[bg * 6 + 3], b4 = sBB[bg * 6 + 4], b5 = sBB[bg * 6 + 5];
            const float tl = x - b0, tt = y - b1, tf = z - b2;
            const float tr = b3 - x, tb = b4 - y, ta = b5 - z;

            const float c1 = fminf(tl, tr) / fmaxf(tl, tr);
            const float c2 = fminf(tt, tb) / fmaxf(tt, tb);
            const float c3 = fminf(tf, ta) / fmaxf(tf, ta);
            const float ctr_t = sqrtf(fminf(fmaxf(c1 * c2 * c3, 1e-8f), 1.0f));

            const float p0 = bp[0], p1 = bp[1], p2 = bp[2], p3 = bp[3], p4 = bp[4], p5 = bp[5];
            const float pv = (p0 + p3) * (p1 + p4) * (p2 + p5);
            const float tv = (tl + tr) * (tt + tb) * (tf + ta);
            const float m0 = fminf(p0, tl), m1 = fminf(p1, tt), m2 = fminf(p2, tf);
            const float m3 = fminf(p3, tr), m4 = fminf(p4, tb), m5 = fminf(p5, ta);
            const float inter = (m0 + m3) * (m1 + m4) * (m2 + m5);
            const float uni   = pv + tv - inter;
            const float iou   = (inter + 1.0f) / (uni + 1.0f);
            const float il    = -logf(fmaxf(iou, 1e-6f));
            s_box = il * ctr_t;
            s_w   = ctr_t;

            const float ct  = ctr_pred[(size_t)b * N + n];
            s_ctr = fmaxf(ct, 0.0f) - ct * ctr_t + log1pf(expf(-fabsf(ct)));
        }
    }

    // ---- wave32 reductions via WMMA (all lanes active; EXEC == all-1s) ----
    float r0 = wave_sum32(s_cls);
    float r1 = wave_sum32(s_npos);
    float r2 = wave_sum32(s_box);
    float r3 = wave_sum32(s_w);
    float r4 = wave_sum32(s_ctr);

    const int wave = tid >> 5;
    if ((tid & 31) == 0) {
        sRed[wave][0] = r0; sRed[wave][1] = r1; sRed[wave][2] = r2;
        sRed[wave][3] = r3; sRed[wave][4] = r4;
    }
    __syncthreads();
    if (tid == 0) {
        float a0 = 0, a1 = 0, a2 = 0, a3 = 0, a4 = 0;
        const int nw = blockDim.x >> 5;
        for (int w = 0; w < nw; ++w) {
            a0 += sRed[w][0]; a1 += sRed[w][1]; a2 += sRed[w][2];
            a3 += sRed[w][3]; a4 += sRed[w][4];
        }
        const size_t blk = (size_t)blockIdx.y * gridDim.x + blockIdx.x;
        float* o = partials + blk * 5;
        o[0] = a0; o[1] = a1; o[2] = a2; o[3] = a3; o[4] = a4;
    }
}

__global__ void fcos_finalize_kernel(const float* __restrict__ partials,
                                     int nblk, int B, float* __restrict__ out)
{
    const int lane = threadIdx.x;   // 32 threads, one wave
    float a0 = 0, a1 = 0, a2 = 0, a3 = 0, a4 = 0;
    for (int i = lane; i < nblk; i += 32) {
        const float* p = partials + (size_t)i * 5;
        a0 += p[0]; a1 += p[1]; a2 += p[2]; a3 += p[3]; a4 += p[4];
    }
    a0 = wave_sum32(a0);
    a1 = wave_sum32(a1);
    a2 = wave_sum32(a2);
    a3 = wave_sum32(a3);
    a4 = wave_sum32(a4);
    if (lane == 0) {
        const float num_pos = a1;
        out[0] = a0 / (num_pos + (float)B);          // cls_loss
        out[1] = a2 / fmaxf(a3, 1e-8f);              // box_loss
        out[2] = a4 / fmaxf(num_pos, 1.0f);          // center_loss
    }
}

extern "C" void kernel_launch(void* const* d_in, const int* in_sizes, int n_in,
                              void* d_out, int out_size, void* d_ws, size_t ws_size,
                              hipStream_t stream) {
    const float* loc   = (const float*)d_in[0];  // locations  (N,3)
    const float* cls   = (const float*)d_in[1];  // cls_pred   (B,N,C)
    const float* boxp  = (const float*)d_in[2];  // box_pred   (B,N,6)
    const float* ctrp  = (const float*)d_in[3];  // center_pred(B,N)
    const float* bbx   = (const float*)d_in[4];  // bboxes     (B,G,6)
    const int*   glab  = (const int*)d_in[5];    // gt_labels  (B,G)
    // d_in[6] = gt_centers, unused (centers recomputed from bboxes as in reference)
    const float* sptp  = (const float*)d_in[7];  // strides_pt (N)
    const float* soip  = (const float*)d_in[8];  // soi        (N,2)

    const int N  = in_sizes[0] / 3;
    const int BN = in_sizes[3];
    const int B  = BN / N;
    const int G  = in_sizes[5] / B;
    const int C  = in_sizes[1] / BN;

    const int bx   = (N + 255) / 256;
    const int nblk = bx * B;
    float* partials = (float*)d_ws;

    dim3 grid(bx, B);
    fcos_fused_kernel<<<grid, 256, 0, stream>>>(loc, cls, boxp, ctrp, bbx, glab,
                                                sptp, soip, partials, N, B, G, C);
    fcos_finalize_kernel<<<1, 32, 0, stream>>>(partials, nblk, B, (float*)d_out);
}